// LIFCell_55645596287160
// MI455X (gfx1250) — compile-verified
//
#include <hip/hip_runtime.h>
#include <cstdint>
#include <cstddef>

// ---------------------------------------------------------------------------
// LIF scan over time, CDNA5 (gfx1250).
//   v = v*sigmoid(decay)*(1-z) + x[:,t,:];  z = (v > 0.5);  out[:,t,:] = z
// Streaming-bound: 537 MB @ 23.3 TB/s => ~23 us floor. No matmul => no WMMA;
// the CDNA5 feature that matters is the async global->LDS DMA (ASYNCcnt).
//
// This version makes every wave fully self-sufficient: a wave DMAs exactly
// the 4 KB sub-tile it will read, so the ONLY sync is its own
// s_wait_asynccnt — zero workgroup barriers, waves never convoy.
// Pipeline depth 8 => 7 tiles (112 KB/block, ~28 MB machine-wide) in flight,
// matching the 23.3 TB/s x ~1 us bandwidth-delay product.
// ---------------------------------------------------------------------------

typedef float v4f __attribute__((ext_vector_type(4)));

#define B_       512
#define T_       512
#define H_       256
#define ROWS     2                      // batch rows per block
#define TILE_T   8                      // timesteps per tile
#define NTHREADS 128                    // 4 wave32
#define QD       8                      // pipeline depth (LDS buffers)
#define NT       (T_ / TILE_T)          // 64 tiles
#define TILE_FLOATS (ROWS * TILE_T * H_)   // 4096 floats = 16 KB / tile
#define ASYNC_PER_WAVE TILE_T              // 8 async b128 per wave per tile

static_assert(ASYNC_PER_WAVE == 8, "wait literals assume 8 asyncs/tile/wave");
static_assert(QD == 8, "buffer indexing + wait chain assume depth 8");
static_assert((QD - 1) * ASYNC_PER_WAVE <= 63, "ASYNCcnt is 6 bits");

// One lane copies 16 contiguous bytes global -> LDS via the async DMA path.
// VDST VGPR = LDS byte offset (low 32 bits of a flat LDS pointer, ISA 10.2),
// VADDR = 64-bit global address, GV mode (saddr = off). Tracked by ASYNCcnt.
__device__ __forceinline__ void async_b128(uint32_t lds_byte_off, const void* gaddr) {
  asm volatile("global_load_async_to_lds_b128 %0, %1, off"
               :: "v"(lds_byte_off), "v"(gaddr)
               : "memory");
}

#define WAIT_ASYNCCNT(n) asm volatile("s_wait_asynccnt " #n ::: "memory")

// Wave-private tile DMA: 8 instructions, each wave32 x 16B = 512B (one
// timestep's 128-float h-segment), 4 KB total = exactly what this wave reads.
__device__ __forceinline__ void issue_tile_wave(const float* __restrict__ xrow, // x + brow*T*H + hseg + lane4
                                                int t0, uint32_t lds_wave_base)
{
#pragma unroll
  for (int p = 0; p < TILE_T; ++p) {
    async_b128(lds_wave_base + (uint32_t)(p * H_ * 4),
               xrow + (size_t)(t0 + p) * H_);
  }
}

__global__ __launch_bounds__(NTHREADS)
void lif_scan_kernel(const float* __restrict__ x,      // [B,T,H]
                     const float* __restrict__ decay,  // [H]
                     const float* __restrict__ v0,     // [B,H]
                     const float* __restrict__ z0,     // [B,H]
                     float* __restrict__ out)          // [B,T,H]
{
  __shared__ float buf[QD][TILE_FLOATS];               // 128 KB

  const int tid   = threadIdx.x;
  const int lane4 = (tid & 31) << 2;        // 4 consecutive h per lane
  const int wv    = tid >> 5;               // wave id 0..3
  const int bsub  = wv >> 1;                // 0..ROWS-1
  const int hseg  = (wv & 1) << 7;          // 0 or 128
  const int h0    = hseg + lane4;
  const int brow  = blockIdx.x * ROWS + bsub;

  // Per-lane source pointer for this wave's h-segment of its batch row.
  const float* xrow = x + (size_t)brow * T_ * H_ + h0;

  // Wave-private LDS byte offsets (flat LDS ptr low 32 bits == LDS offset).
  const uint32_t wave_off = (uint32_t)((bsub * TILE_T * H_ + h0) * sizeof(float));
  uint32_t lds_base[QD];
#pragma unroll
  for (int q = 0; q < QD; ++q)
    lds_base[q] = (uint32_t)(uintptr_t)(&buf[q][0]) + wave_off;

  // ---- per-thread parameters / state (4 h-lanes each) ----
  float d[4], v[4], z[4];
  {
    v4f dv = *(const v4f*)(decay + h0);
    v4f vv = *(const v4f*)(v0 + (size_t)brow * H_ + h0);
    v4f zv = *(const v4f*)(z0 + (size_t)brow * H_ + h0);
#pragma unroll
    for (int i = 0; i < 4; ++i) {
      d[i] = 1.0f / (1.0f + __expf(-dv[i]));           // sigmoid(decay), once
      v[i] = vv[i];
      z[i] = zv[i];
    }
  }

  // ---- prologue: prefetch first QD-1 tiles (56 asyncs outstanding max) ----
#pragma unroll
  for (int q = 0; q < QD - 1; ++q)
    issue_tile_wave(xrow, q * TILE_T, lds_base[q]);

  for (int tile = 0; tile < NT; ++tile) {
    const int t0 = tile * TILE_T;

    // Wait until at most rem*8 asyncs remain outstanding.
    // ASYNCcnt decrements in issue order, so that covers this tile's 8 DMAs only when rem < 7 (tile >= 57).
    const int rem = (NT - 1 - tile) < (QD - 1) ? (NT - 1 - tile) : (QD - 1);
    switch (rem) {
      case 7:  WAIT_ASYNCCNT(56); break;
      case 6:  WAIT_ASYNCCNT(48); break;
      case 5:  WAIT_ASYNCCNT(40); break;
      case 4:  WAIT_ASYNCCNT(32); break;
      case 3:  WAIT_ASYNCCNT(24); break;
      case 2:  WAIT_ASYNCCNT(16); break;
      case 1:  WAIT_ASYNCCNT(8);  break;
      default: WAIT_ASYNCCNT(0);  break;
    }

    const float* src = &buf[tile & (QD - 1)][bsub * TILE_T * H_ + h0];
    float*       dst = out + ((size_t)brow * T_ + (size_t)t0) * H_ + h0;

#pragma unroll
    for (int tl = 0; tl < TILE_T; ++tl) {
      v4f xv = *(const v4f*)(src + tl * H_);           // ds_load_b128 (own DMA)
      v4f zo;
#pragma unroll
      for (int i = 0; i < 4; ++i) {
        float vn = fmaf(v[i] * d[i], 1.0f - z[i], xv[i]);
        float zn = vn > 0.5f ? 1.0f : 0.0f;            // spike threshold
        v[i] = vn;
        z[i] = zn;
        zo[i] = zn;
      }
      // write-once stream: non-temporal b128 store, keep L2 for the x stream
      __builtin_nontemporal_store(zo, (v4f*)(dst + (size_t)tl * H_));
    }

    // Refill the buffer drained in the previous iteration (wave-private region; ordered by
    // program order after its last ds_load, and by ASYNCcnt thereafter).
    const int ahead = tile + (QD - 1);
    if (ahead < NT)
      issue_tile_wave(xrow, ahead * TILE_T, lds_base[ahead & (QD - 1)]);
  }
}

extern "C" void kernel_launch(void* const* d_in, const int* in_sizes, int n_in,
                              void* d_out, int out_size, void* d_ws, size_t ws_size,
                              hipStream_t stream) {
  const float* x     = (const float*)d_in[0];
  const float* decay = (const float*)d_in[1];
  const float* v0    = (const float*)d_in[2];
  const float* z0    = (const float*)d_in[3];
  float* out = (float*)d_out;

  dim3 grid(B_ / ROWS);     // 256 blocks
  dim3 block(NTHREADS);     // 128 threads = 4 wave32, fully decoupled waves
  hipLaunchKernelGGL(lif_scan_kernel, grid, block, 0, stream,
                     x, decay, v0, z0, out);
}